// GCL2D_22084721836286
// MI455X (gfx1250) — compile-verified
//
#include <hip/hip_runtime.h>
#include <math.h>

// ---- CDNA5 WMMA types ----
typedef float v2f __attribute__((ext_vector_type(2)));
typedef float v8f __attribute__((ext_vector_type(8)));

#define HW     16384      // 128*128
#define HWT    163840     // 128*128*10

__device__ __forceinline__ float block_reduce_sum(float v, float* sm) {
    int t = threadIdx.x;
    sm[t] = v; __syncthreads();
    for (int s = 128; s > 0; s >>= 1) {
        if (t < s) sm[t] += sm[t + s];
        __syncthreads();
    }
    float r = sm[0];
    __syncthreads();
    return r;
}

// Kernel 1: per-field stencil y = z + adv + diff, pack z/u8/u9 contiguously,
// and reduce squared norms ny, nz, na, nb.
__global__ void GCL2D_stencil_norms(const float* __restrict__ x1,
                                    const float* __restrict__ x2,
                                    const float* __restrict__ vs,
                                    const float* __restrict__ u,
                                    const float* __restrict__ dt,
                                    const float* __restrict__ dxp,
                                    float* __restrict__ y,
                                    float* __restrict__ z,
                                    float* __restrict__ pu8,
                                    float* __restrict__ pu9,
                                    float* __restrict__ ny,
                                    float* __restrict__ nz,
                                    float* __restrict__ na,
                                    float* __restrict__ nb) {
    const int i  = blockIdx.x;        // 0..31 over reps = concat(x2, x1)
    const int bi = i & 15;
    const float dx  = dxp[0];
    const float c   = vs[bi * 3 + 0];
    const float cx  = vs[bi * 3 + 1];
    const float cy  = vs[bi * 3 + 2];
    const float dtv = dt[bi];
    const float inv_dx  = 1.0f / dx;
    const float inv_dx2 = inv_dx * inv_dx;
    const float* f = (i < 16) ? (x2 + (size_t)i * HWT) : (x1 + (size_t)(i - 16) * HWT);

    float sy = 0.f, sz = 0.f, sa = 0.f, sb = 0.f;
    for (int p = threadIdx.x; p < HW; p += 256) {
        int h = p >> 7, w = p & 127;
        float u0  = f[(size_t)p * 10];
        float uxp = f[(size_t)(h * 128 + ((w + 1) & 127)) * 10]; // roll -1, axis=2
        float uxm = f[(size_t)(h * 128 + ((w - 1) & 127)) * 10]; // roll +1, axis=2
        float uyp = f[(size_t)((((h + 1) & 127) * 128) + w) * 10]; // roll -1, axis=1
        float uym = f[(size_t)((((h - 1) & 127) * 128) + w) * 10]; // roll +1, axis=1

        float dw_m = u0 - uxp;
        float dw_p = u0 - uxm;
        float dh_p = u0 - uym;
        float dh_m = u0 - uyp;
        float xterm = (cx <= 0.0f) ? (-cx * dw_m) : (cx * dw_p);
        float yterm = (cy >= 0.0f) ? (cy * dh_p) : (-cy * dh_m);
        float adv   = -dtv * (u0 * (xterm + yterm)) * inv_dx;
        float lap   = uym + uyp + uxm + uxp - 4.0f * u0;
        float dif   = c * lap * inv_dx2;
        float yy    = u0 + adv + dif;

        y[(size_t)i * HW + p] = yy;
        z[(size_t)i * HW + p] = u0;
        sy += yy * yy;
        sz += u0 * u0;
        if (i < 16) {
            float a = u[(size_t)i * HWT + (size_t)p * 10 + 8];
            float b = u[(size_t)i * HWT + (size_t)p * 10 + 9];
            pu8[(size_t)i * HW + p] = a;
            pu9[(size_t)i * HW + p] = b;
            sa += a * a;
            sb += b * b;
        }
    }

    __shared__ float sm[256];
    sy = block_reduce_sum(sy, sm);
    sz = block_reduce_sum(sz, sm);
    sa = block_reduce_sum(sa, sm);
    sb = block_reduce_sum(sb, sm);
    if (threadIdx.x == 0) {
        ny[i] = sy;
        nz[i] = sz;
        if (i < 16) { na[i] = sa; nb[i] = sb; }
    }
}

// Kernel 2: Gram matrices via V_WMMA_F32_16X16X4_F32.
// Jobs 0..3: 16x16 tiles of Gpd[i][j] = <y_i, z_j> (32x32).
// Job 4:     Gus[i][j] = <u8_i, u9_j> (16x16).
// 8 waves split K=16384 into slices of 2048 (512 WMMAs each); LDS reduce.
__global__ void GCL2D_gram_wmma(const float* __restrict__ Ay,
                                const float* __restrict__ Bz,
                                const float* __restrict__ Au8,
                                const float* __restrict__ Bu9,
                                float* __restrict__ Gpd,
                                float* __restrict__ Gus) {
    const int job = blockIdx.x;
    const float* A;
    const float* B;
    if (job < 4) {
        int ti = job >> 1, tj = job & 1;
        A = Ay + (size_t)ti * 16 * HW;
        B = Bz + (size_t)tj * 16 * HW;
    } else {
        A = Au8;
        B = Bu9;
    }

    const int lane = threadIdx.x & 31;
    const int wave = threadIdx.x >> 5;
    const int row  = lane & 15;            // A: M row / B: N column (both row-major, K contiguous)
    const int kh   = (lane >> 4) << 1;     // K sub-offset per ISA 16x4 f32 layout
    const float* ap = A + (size_t)row * HW + wave * 2048 + kh;
    const float* bp = B + (size_t)row * HW + wave * 2048 + kh;

    v8f acc = {0.f, 0.f, 0.f, 0.f, 0.f, 0.f, 0.f, 0.f};
    for (int it = 0; it < 512; ++it) {
        v2f a = *(const v2f*)ap;
        v2f b = *(const v2f*)bp;
        acc = __builtin_amdgcn_wmma_f32_16x16x4_f32(
            false, a, false, b, (short)0, acc, false, false);
        ap += 4;
        bp += 4;
    }

    __shared__ float red[8 * 256];
#pragma unroll
    for (int r = 0; r < 8; ++r) red[wave * 256 + lane * 8 + r] = acc[r];
    __syncthreads();

    const int t  = threadIdx.x;
    const int rr = t >> 5;   // VGPR index 0..7
    const int ll = t & 31;   // lane 0..31
    float s = 0.f;
#pragma unroll
    for (int w = 0; w < 8; ++w) s += red[w * 256 + ll * 8 + rr];
    const int M = rr + ((ll >> 4) << 3); // C/D layout: M = vgpr + 8*(lane>=16)
    const int N = ll & 15;
    if (job < 4) {
        int ti = job >> 1, tj = job & 1;
        Gpd[(ti * 16 + M) * 32 + (tj * 16 + N)] = s;
    } else {
        Gus[M * 16 + N] = s;
    }
}

// Kernel 3: assemble us, pd, sim, sim_score; single-pass P / Q / min(sim^2);
// out = (P + Q * max(TAU - min, 0)) / 1024 / 16.
__global__ void GCL2D_finalize(const float* __restrict__ vs,
                               const float* __restrict__ Gpd,
                               const float* __restrict__ Gus,
                               const float* __restrict__ ny,
                               const float* __restrict__ nz,
                               const float* __restrict__ na,
                               const float* __restrict__ nb,
                               float* __restrict__ out) {
    __shared__ float sP[256], sQ[256], sM[256];
    float P = 0.f, Q = 0.f, mn = 3.402823466e38f;
    for (int e = threadIdx.x; e < 1024; e += 256) {
        int ii = e >> 5, jj = e & 31;
        int bi = ii & 15, bj = jj & 15;

        float usq = na[bi] + nb[bj] - 2.0f * Gus[bi * 16 + bj];
        float uss = sqrtf(fmaxf(usq, 0.0f));
        float pdq = ny[ii] + nz[jj] - 2.0f * Gpd[ii * 32 + jj];
        float pdd = sqrtf(fmaxf(pdq, 0.0f));
        float d   = uss - pdd;
        float sim = d * d;       // (us - pd)^2
        float s2  = sim * sim;   // sim^2

        float t0i = vs[bi * 3];
        float t1i = sqrtf(vs[bi * 3 + 1] * vs[bi * 3 + 1] + vs[bi * 3 + 2] * vs[bi * 3 + 2]);
        float t0j = vs[bj * 3];
        float t1j = sqrtf(vs[bj * 3 + 1] * vs[bj * 3 + 1] + vs[bj * 3 + 2] * vs[bj * 3 + 2]);
        float nvi  = sqrtf(t0i * t0i + t1i * t1i);
        float nvj  = sqrtf(t0j * t0j + t1j * t1j);
        float prod = sqrtf(fabsf(t0i * t0j) + fabsf(t1i * t1j));
        float score = prod / fmaxf(nvi, nvj);

        P  += 0.5f * score * s2;
        Q  += 0.5f * (1.0f - score);
        mn = fminf(mn, s2);
    }
    int t = threadIdx.x;
    sP[t] = P; sQ[t] = Q; sM[t] = mn;
    __syncthreads();
    for (int s = 128; s > 0; s >>= 1) {
        if (t < s) {
            sP[t] += sP[t + s];
            sQ[t] += sQ[t + s];
            sM[t] = fminf(sM[t], sM[t + s]);
        }
        __syncthreads();
    }
    if (t == 0) {
        float scal = fmaxf(100.0f - sM[0], 0.0f);  // TAU = 100
        out[0] = (sP[0] + sQ[0] * scal) * (1.0f / 1024.0f) * (1.0f / 16.0f);
    }
}

extern "C" void kernel_launch(void* const* d_in, const int* in_sizes, int n_in,
                              void* d_out, int out_size, void* d_ws, size_t ws_size,
                              hipStream_t stream) {
    const float* x1 = (const float*)d_in[0];
    const float* x2 = (const float*)d_in[1];
    const float* vs = (const float*)d_in[2];
    const float* u  = (const float*)d_in[3];
    const float* dt = (const float*)d_in[4];
    const float* dx = (const float*)d_in[5];
    float* out = (float*)d_out;

    // Workspace layout (floats): y[32*HW] z[32*HW] u8[16*HW] u9[16*HW]
    //                            Gpd[32*32] Gus[16*16] ny[32] nz[32] na[16] nb[16]
    float* W   = (float*)d_ws;
    float* y   = W;
    float* z   = W + 32 * HW;                 //  524288
    float* pu8 = W + 64 * HW;                 // 1048576
    float* pu9 = W + 80 * HW;                 // 1310720
    float* Gpd = W + 96 * HW;                 // 1572864
    float* Gus = Gpd + 32 * 32;
    float* ny  = Gus + 16 * 16;
    float* nz  = ny + 32;
    float* na  = nz + 32;
    float* nb  = na + 16;

    GCL2D_stencil_norms<<<32, 256, 0, stream>>>(x1, x2, vs, u, dt, dx,
                                                y, z, pu8, pu9, ny, nz, na, nb);
    GCL2D_gram_wmma<<<5, 256, 0, stream>>>(y, z, pu8, pu9, Gpd, Gus);
    GCL2D_finalize<<<1, 256, 0, stream>>>(vs, Gpd, Gus, ny, nz, na, nb, out);
}